// BiMambaBlock_11450382811523
// MI455X (gfx1250) — compile-verified
//
#include <hip/hip_runtime.h>

// ---------------- problem dimensions (compile-time) ----------------
#define D_MODEL  768
#define D_STATE  16
#define D_CONVW  4
#define D_INNER  1536
#define DT_RANK  48
#define BATCH    2
#define SEQ      1024
#define NROWS    (BATCH * SEQ)        // 2048
#define XZ_COLS  (2 * D_INNER)        // 3072
#define DBC_COLS (DT_RANK + 2 * D_STATE) // 80

typedef float v2f __attribute__((ext_vector_type(2)));
typedef float v8f __attribute__((ext_vector_type(8)));

// f32 WMMA: D(16x16,f32) = A(16x4,f32) x B(4x16,f32) + C
__device__ __forceinline__ v8f wmma_k4(v2f a, v2f b, v8f c) {
  return __builtin_amdgcn_wmma_f32_16x16x4_f32(false, a, false, b, (short)0, c,
                                               false, false);
}

__device__ __forceinline__ float silu_f(float z) {
  return z * (1.0f / (1.0f + __expf(-z)));
}

// Multi-tile WMMA K-loop: one 16-row A strip vs CT 16-col B tiles.
// CT independent accumulators -> no WMMA->WMMA RAW chain inside a K-step.
template <int CT>
__device__ __forceinline__ void gemm_tiles(const float* __restrict__ Ap,
                                           const float* __restrict__ Bp,
                                           size_t tstride, int K,
                                           v8f (&acc)[CT]) {
  for (int k0 = 0; k0 < K; k0 += 32) {
    __builtin_prefetch(Ap + k0 + 512);
#pragma unroll
    for (int t = 0; t < CT; ++t)
      __builtin_prefetch(Bp + (size_t)t * tstride + k0 + 512);
#pragma unroll
    for (int kk = 0; kk < 32; kk += 4) {
      const int k = k0 + kk;
      v2f a = *(const v2f*)(Ap + k);
#pragma unroll
      for (int t = 0; t < CT; ++t) {
        v2f b = *(const v2f*)(Bp + (size_t)t * tstride + k);
        acc[t] = wmma_k4(a, b, acc[t]);
      }
    }
  }
}

// Store CT 16x16 f32 C/D tiles: VGPR v holds row (v + 8*half), col = lane&15.
template <int CT>
__device__ __forceinline__ void store_tiles(float* __restrict__ out, size_t ldc,
                                            const v8f (&acc)[CT], int half) {
#pragma unroll
  for (int t = 0; t < CT; ++t)
#pragma unroll
    for (int v = 0; v < 8; ++v)
      out[(size_t)(v + 8 * half) * ldc + 16 * t] = acc[t][v];
}

// ---------------- 1. LayerNorm ----------------
__global__ __launch_bounds__(256) void ln_kernel(const float* __restrict__ x,
                                                 const float* __restrict__ g,
                                                 const float* __restrict__ b,
                                                 float* __restrict__ x0) {
  __shared__ float s_sum[256];
  __shared__ float s_sq[256];
  const int r = blockIdx.x;
  const float* xr = x + (size_t)r * D_MODEL;
  float lsum = 0.f, lsq = 0.f;
  for (int c = threadIdx.x; c < D_MODEL; c += 256) {
    float v = xr[c];
    lsum += v;
    lsq += v * v;
  }
  s_sum[threadIdx.x] = lsum;
  s_sq[threadIdx.x] = lsq;
  __syncthreads();
  for (int s = 128; s > 0; s >>= 1) {
    if ((int)threadIdx.x < s) {
      s_sum[threadIdx.x] += s_sum[threadIdx.x + s];
      s_sq[threadIdx.x] += s_sq[threadIdx.x + s];
    }
    __syncthreads();
  }
  const float mean = s_sum[0] * (1.0f / D_MODEL);
  const float var = s_sq[0] * (1.0f / D_MODEL) - mean * mean;
  const float inv = rsqrtf(var + 1e-5f);
  for (int c = threadIdx.x; c < D_MODEL; c += 256) {
    x0[(size_t)r * D_MODEL + c] = (xr[c] - mean) * inv * g[c] + b[c];
  }
}

// ---------------- 2. xz = x0 @ in_w^T (both dirs; dir1 reads flipped rows) --
// Wave computes a 16x64 strip (4 col tiles).
__global__ __launch_bounds__(128) void gemm_in_kernel(
    const float* __restrict__ x0, const float* __restrict__ wf,
    const float* __restrict__ wb, float* __restrict__ xz) {
  const int lane = threadIdx.x & 31;
  const int wave = threadIdx.x >> 5;
  const int half = lane >> 4;
  const int ln = lane & 15;
  const int dir = blockIdx.z;
  const int r0 = blockIdx.y * 64 + wave * 16;
  const int c0 = blockIdx.x * 64;

  int lrow = r0 + ln;
  int arow = lrow;
  if (dir) {
    int bb = lrow >> 10, l = lrow & 1023;
    arow = (bb << 10) + (1023 - l);
  }
  const float* W = dir ? wb : wf;
  const float* Ap = x0 + (size_t)arow * D_MODEL + 2 * half;
  const float* Bp = W + (size_t)(c0 + ln) * D_MODEL + 2 * half;

  v8f acc[4] = {};
  gemm_tiles<4>(Ap, Bp, (size_t)16 * D_MODEL, D_MODEL, acc);

  float* out = xz + ((size_t)dir * NROWS + r0) * XZ_COLS + c0 + ln;
  store_tiles<4>(out, XZ_COLS, acc, half);
}

// ---------------- 3. causal depthwise conv(4) + SiLU -> xc ----------------
__global__ __launch_bounds__(256) void conv_silu_kernel(
    const float* __restrict__ xz, const float* __restrict__ cwf,
    const float* __restrict__ cbf, const float* __restrict__ cwb,
    const float* __restrict__ cbb, float* __restrict__ xc) {
  const size_t total = (size_t)2 * NROWS * D_INNER;
  size_t i = (size_t)blockIdx.x * 256 + threadIdx.x;
  if (i >= total) return;
  const int d = (int)(i % D_INNER);
  const int gr = (int)(i / D_INNER);  // 0..2*NROWS-1, dir-major
  const int dir = gr >= NROWS;
  const int r = gr - dir * NROWS;
  const int l = r & 1023;
  const float* w = (dir ? cwb : cwf) + d * D_CONVW;
  const float bias = (dir ? cbb : cbf)[d];
  const float* col = xz + (size_t)gr * XZ_COLS + d;  // xc half of xz
  float s = bias;
#pragma unroll
  for (int j = 0; j < D_CONVW; ++j) {
    int ls = l - 3 + j;
    if (ls >= 0) s += w[j] * col[(ptrdiff_t)(ls - l) * (ptrdiff_t)XZ_COLS];
  }
  xc[i] = silu_f(s);
}

// ---------------- 4. dbc = xc @ xproj_w^T (wave computes 16x80) -----------
__global__ __launch_bounds__(128) void gemm_xproj_kernel(
    const float* __restrict__ xc, const float* __restrict__ wf,
    const float* __restrict__ wb, float* __restrict__ dbc) {
  const int lane = threadIdx.x & 31;
  const int wave = threadIdx.x >> 5;
  const int half = lane >> 4;
  const int ln = lane & 15;
  const int dir = blockIdx.z;
  const int r0 = blockIdx.x * 64 + wave * 16;

  const float* W = dir ? wb : wf;
  const float* Ap = xc + ((size_t)dir * NROWS + r0 + ln) * D_INNER + 2 * half;
  const float* Bp = W + (size_t)ln * D_INNER + 2 * half;

  v8f acc[5] = {};
  gemm_tiles<5>(Ap, Bp, (size_t)16 * D_INNER, D_INNER, acc);

  float* out = dbc + ((size_t)dir * NROWS + r0) * DBC_COLS + ln;
  store_tiles<5>(out, DBC_COLS, acc, half);
}

// ---------------- 5. selective scan (delta fused) ----------------
// One lane per (dir, batch, channel); (dir,b) derived from blockIdx so the
// per-timestep dt/B/C row pointer is provably scalar (SMEM broadcast loads).
// y overwrites xc in place.
__global__ __launch_bounds__(256) void scan_kernel(
    const float* __restrict__ xz, float* __restrict__ xc,
    const float* __restrict__ dbc, const float* __restrict__ dtw_f,
    const float* __restrict__ dtb_f, const float* __restrict__ alog_f,
    const float* __restrict__ dd_f, const float* __restrict__ dtw_b,
    const float* __restrict__ dtb_b, const float* __restrict__ alog_b,
    const float* __restrict__ dd_b) {
  const int blk = blockIdx.x;                // 24 blocks
  const int dir = blk / 12;                  // uniform
  const int b = (blk % 12) / 6;              // uniform (1536/256 == 6)
  const int d = (blk % 6) * 256 + threadIdx.x;

  const float* dtw = (dir ? dtw_b : dtw_f) + d * DT_RANK;
  const float dtb = (dir ? dtb_b : dtb_f)[d];
  const float* al = (dir ? alog_b : alog_f) + d * D_STATE;
  const float Dv = (dir ? dd_b : dd_f)[d];

  float w[DT_RANK];
#pragma unroll
  for (int j = 0; j < DT_RANK; ++j) w[j] = dtw[j];
  float An[D_STATE], h[D_STATE];
#pragma unroll
  for (int n = 0; n < D_STATE; ++n) {
    An[n] = -__expf(al[n]);
    h[n] = 0.f;
  }

  const int rbase = dir * NROWS + b * SEQ;   // uniform
  const float* rowp = dbc + (size_t)rbase * DBC_COLS;  // scalar pointer
  const float* up = xc + (size_t)rbase * D_INNER + d;
  const float* zp = xz + (size_t)rbase * XZ_COLS + D_INNER + d;
  float* yp = xc + (size_t)rbase * D_INNER + d;

  for (int l = 0; l < SEQ; ++l) {
    // delta = softplus(dt . dt_w[d] + dt_b[d])
    float acc = dtb;
    const float4* dq = (const float4*)rowp;
#pragma unroll
    for (int q = 0; q < DT_RANK / 4; ++q) {
      float4 t = dq[q];
      acc += t.x * w[4 * q] + t.y * w[4 * q + 1] + t.z * w[4 * q + 2] +
             t.w * w[4 * q + 3];
    }
    const float dlt = fmaxf(acc, 0.f) + log1pf(__expf(-fabsf(acc)));

    const float u = *up;
    const float zv = *zp;

    float Bv[D_STATE], Cv[D_STATE];
    const float4* p = (const float4*)(rowp + DT_RANK);
#pragma unroll
    for (int q = 0; q < 4; ++q) {
      float4 t = p[q];
      Bv[4 * q] = t.x; Bv[4 * q + 1] = t.y; Bv[4 * q + 2] = t.z; Bv[4 * q + 3] = t.w;
    }
#pragma unroll
    for (int q = 0; q < 4; ++q) {
      float4 t = p[4 + q];
      Cv[4 * q] = t.x; Cv[4 * q + 1] = t.y; Cv[4 * q + 2] = t.z; Cv[4 * q + 3] = t.w;
    }

    float ydot = 0.f;
#pragma unroll
    for (int n = 0; n < D_STATE; ++n) {
      float hn = __expf(dlt * An[n]) * h[n] + (dlt * Bv[n]) * u;
      h[n] = hn;
      ydot += hn * Cv[n];
    }
    *yp = (ydot + u * Dv) * silu_f(zv);

    rowp += DBC_COLS;
    up += D_INNER;
    zp += XZ_COLS;
    yp += D_INNER;
  }
}

// ---------------- 6. out = x + y_f @ out_w_f^T + flip(y_b @ out_w_b^T) -----
// Wave computes a 16x64 strip of the output.
__global__ __launch_bounds__(128) void gemm_out_kernel(
    const float* __restrict__ y, const float* __restrict__ wf,
    const float* __restrict__ wb, const float* __restrict__ x,
    float* __restrict__ out) {
  const int lane = threadIdx.x & 31;
  const int wave = threadIdx.x >> 5;
  const int half = lane >> 4;
  const int ln = lane & 15;
  const int r0 = blockIdx.y * 64 + wave * 16;
  const int c0 = blockIdx.x * 64;

  const int lrow = r0 + ln;
  const int bb = lrow >> 10, l = lrow & 1023;
  const int brow = (bb << 10) + (1023 - l);

  const float* Af = y + (size_t)lrow * D_INNER + 2 * half;
  const float* Ab = y + ((size_t)NROWS + brow) * D_INNER + 2 * half;
  const float* Bf = wf + (size_t)(c0 + ln) * D_INNER + 2 * half;
  const float* Bb = wb + (size_t)(c0 + ln) * D_INNER + 2 * half;

  v8f acc[4] = {};
  gemm_tiles<4>(Af, Bf, (size_t)16 * D_INNER, D_INNER, acc);
  gemm_tiles<4>(Ab, Bb, (size_t)16 * D_INNER, D_INNER, acc);

  const float* xb = x + (size_t)r0 * D_MODEL + c0 + ln;
  float* ob = out + (size_t)r0 * D_MODEL + c0 + ln;
#pragma unroll
  for (int t = 0; t < 4; ++t)
#pragma unroll
    for (int v = 0; v < 8; ++v) {
      const size_t off = (size_t)(v + 8 * half) * D_MODEL + 16 * t;
      ob[off] = xb[off] + acc[t][v];
    }
}

// ---------------- host launcher ----------------
extern "C" void kernel_launch(void* const* d_in, const int* in_sizes, int n_in,
                              void* d_out, int out_size, void* d_ws,
                              size_t ws_size, hipStream_t stream) {
  (void)in_sizes; (void)n_in; (void)out_size; (void)ws_size;
  const float* x       = (const float*)d_in[0];
  const float* ln_g    = (const float*)d_in[1];
  const float* ln_b    = (const float*)d_in[2];
  const float* f_in_w  = (const float*)d_in[3];
  const float* f_cw    = (const float*)d_in[4];
  const float* f_cb    = (const float*)d_in[5];
  const float* f_xp    = (const float*)d_in[6];
  const float* f_dtw   = (const float*)d_in[7];
  const float* f_dtb   = (const float*)d_in[8];
  const float* f_alog  = (const float*)d_in[9];
  const float* f_D     = (const float*)d_in[10];
  const float* f_outw  = (const float*)d_in[11];
  const float* b_in_w  = (const float*)d_in[12];
  const float* b_cw    = (const float*)d_in[13];
  const float* b_cb    = (const float*)d_in[14];
  const float* b_xp    = (const float*)d_in[15];
  const float* b_dtw   = (const float*)d_in[16];
  const float* b_dtb   = (const float*)d_in[17];
  const float* b_alog  = (const float*)d_in[18];
  const float* b_D     = (const float*)d_in[19];
  const float* b_outw  = (const float*)d_in[20];

  float* ws  = (float*)d_ws;
  float* x0  = ws;                                     // 2048*768
  float* xz  = x0 + (size_t)NROWS * D_MODEL;           // 2*2048*3072
  float* xc  = xz + (size_t)2 * NROWS * XZ_COLS;       // 2*2048*1536 (reused as y)
  float* dbc = xc + (size_t)2 * NROWS * D_INNER;       // 2*2048*80
  float* out = (float*)d_out;

  ln_kernel<<<NROWS, 256, 0, stream>>>(x, ln_g, ln_b, x0);

  gemm_in_kernel<<<dim3(XZ_COLS / 64, NROWS / 64, 2), 128, 0, stream>>>(
      x0, f_in_w, b_in_w, xz);

  const int conv_total = 2 * NROWS * D_INNER;
  conv_silu_kernel<<<conv_total / 256, 256, 0, stream>>>(xz, f_cw, f_cb, b_cw,
                                                         b_cb, xc);

  gemm_xproj_kernel<<<dim3(NROWS / 64, 1, 2), 128, 0, stream>>>(xc, f_xp, b_xp,
                                                                dbc);

  scan_kernel<<<(2 * BATCH * D_INNER) / 256, 256, 0, stream>>>(
      xz, xc, dbc, f_dtw, f_dtb, f_alog, f_D, b_dtw, b_dtb, b_alog, b_D);

  gemm_out_kernel<<<dim3(D_MODEL / 64, NROWS / 64), 128, 0, stream>>>(
      xc, f_outw, b_outw, x, out);
}